// Model_15307263443706
// MI455X (gfx1250) — compile-verified
//
#include <hip/hip_runtime.h>

// Cloud-in-Cell deposition on MI455X (gfx1250).
//
// Roofline: 8.39M particles * 12B = 100MB stream (4.3us @ 23.3TB/s), but the
// naive path needs 33.5M random global f32 atomics -> L2 atomic-op bound
// (~1 RMW/slice/clk => tens of microseconds). CDNA5 fix: bin particles into
// 64x64 grid tiles (one global u32 atomic per block-tile range reservation,
// ~0.5M total), then accumulate each tile in 320KB-WGP LDS with native
// ds_add_f32 (33.5M adds -> LDS atomic units, massive aggregate rate), and
// merge only the 1-cell halo rings with global_atomic_add_f32 scope:SCOPE_DEV
// (~65K atomics). Binned AoS (134MB) stays resident in the 192MB L2 between
// the two passes. Falls back to the direct-atomic kernel if ws is too small.

#define CIC_GRID 1024
#define TILE_BITS 6
#define TILE 64
#define NBIN 256            // 16 x 16 tiles
#define TPAD (TILE + 1)     // 65x65 LDS tile (deposits reach cx+1, cy+1)
#define K3_THREADS 1024
#define K3_PPT 4            // particles per thread per ranking round
#define K4_THREADS 1024
#define WS_HDR 4096         // cursor[256] lives here; binned array after

typedef float f32x2 __attribute__((ext_vector_type(2)));
typedef float f32x4 __attribute__((ext_vector_type(4)));

// Native device-scope global f32 atomic add (RMW resolves in L2 atomic units).
__device__ __forceinline__ void global_fadd_dev(float* base, unsigned byte_off, float val) {
  asm volatile("global_atomic_add_f32 %0, %1, %2 scope:SCOPE_DEV"
               :: "v"(byte_off), "v"(val), "s"(base)
               : "memory");
}

// Native LDS f32 atomic add. Generic pointer to __shared__ carries the LDS
// byte offset in its low 32 bits (DS unit adds LDS_BASE itself).
__device__ __forceinline__ void lds_fadd(float* p, float val) {
  unsigned a = (unsigned)(size_t)p;
  asm volatile("ds_add_f32 %0, %1" :: "v"(a), "v"(val) : "memory");
}

__device__ __forceinline__ void clamp_cell(float x, int& c, float& f) {
  float fl = floorf(x);
  f = x - fl;                       // fraction from UNclamped floor (ref semantics)
  int ci = (int)fl;
  c = ci < 0 ? 0 : (ci > CIC_GRID - 2 ? CIC_GRID - 2 : ci);
}

// ---------------------------------------------------------------------------
// K0: zero the (poisoned) output grid every call + zero the tile cursors.
__global__ __launch_bounds__(256) void zero_kernel(f32x4* __restrict__ out4, int n4,
                                                   unsigned* __restrict__ cursor) {
  int i = blockIdx.x * blockDim.x + threadIdx.x;
  if (i < n4) {
    f32x4 z = {0.0f, 0.0f, 0.0f, 0.0f};
    out4[i] = z;  // global_store_b128
  }
  if (cursor && i < NBIN) cursor[i] = 0u;
}

// ---------------------------------------------------------------------------
// K1: rank-and-scatter particles into per-tile segments of the binned array.
// LDS histogram gives each particle a local rank (ds_add_rtn_u32); one global
// u32 atomic per (block, non-empty tile) reserves the output range.
__global__ __launch_bounds__(K3_THREADS) void scatter_kernel(
    const f32x2* __restrict__ pos, const float* __restrict__ chg,
    f32x4* __restrict__ binned, unsigned* __restrict__ cursor,
    int n, unsigned cap) {
  __shared__ unsigned hist[NBIN];
  __shared__ unsigned base_l[NBIN];
  const int tid = threadIdx.x;
  if (tid < NBIN) hist[tid] = 0u;
  __syncthreads();

  const int i0 = blockIdx.x * (K3_THREADS * K3_PPT) + tid;
  float px[K3_PPT], py[K3_PPT], qv[K3_PPT];
  unsigned tb[K3_PPT], lr[K3_PPT];
  bool ok[K3_PPT];

#pragma unroll
  for (int k = 0; k < K3_PPT; k++) {
    int i = i0 + k * K3_THREADS;
    ok[k] = (i < n);
    if (ok[k]) {
      // Read-once stream: non-temporal so the binned array stays L2-resident.
      f32x2 p = __builtin_nontemporal_load(pos + i);
      float q = __builtin_nontemporal_load(chg + i);
      px[k] = p.x; py[k] = p.y; qv[k] = q;
      int cx, cy; float fx, fy;
      clamp_cell(p.x, cx, fx);
      clamp_cell(p.y, cy, fy);
      tb[k] = ((unsigned)(cx >> TILE_BITS) << 4) | (unsigned)(cy >> TILE_BITS);
      lr[k] = atomicAdd(&hist[tb[k]], 1u);        // ds_add_rtn_u32, local rank
    }
  }
  __syncthreads();

  if (tid < NBIN) {
    unsigned c = hist[tid];
    base_l[tid] = c ? atomicAdd(cursor + tid, c) : 0u;  // one global atomic / tile
  }
  __syncthreads();

#pragma unroll
  for (int k = 0; k < K3_PPT; k++) {
    if (ok[k]) {
      unsigned slot = base_l[tb[k]] + lr[k];
      if (slot < cap) {                            // safety: never OOB
        f32x4 v = {px[k], py[k], qv[k], 0.0f};
        binned[(size_t)tb[k] * cap + slot] = v;    // global_store_b128 (RT -> L2)
      }
    }
  }
}

// ---------------------------------------------------------------------------
// K2: one block per 64x64 tile. Accumulate in a 65x65 LDS tile with native
// ds_add_f32, then write interior cells with plain stores and merge the
// shared halo ring with device-scope global f32 atomics.
__global__ __launch_bounds__(K4_THREADS) void deposit_tile_kernel(
    const f32x4* __restrict__ binned, const unsigned* __restrict__ cursor,
    float* __restrict__ out, unsigned cap) {
  __shared__ float tile[TPAD * TPAD];  // 65*65*4 = 16.9 KB of the 320 KB WGP LDS
  const int t = blockIdx.x;
  const int tid = threadIdx.x;

  for (int idx = tid; idx < TPAD * TPAD; idx += K4_THREADS) tile[idx] = 0.0f;
  __syncthreads();

  const int x0 = (t >> 4) << TILE_BITS;
  const int y0 = (t & 15) << TILE_BITS;
  unsigned cnt = cursor[t];
  if (cnt > cap) cnt = cap;
  const f32x4* seg = binned + (size_t)t * cap;

  for (unsigned i = tid; i < cnt; i += K4_THREADS) {
    f32x4 v = __builtin_nontemporal_load(seg + i);  // read-once (mostly L2 hits)
    int cx, cy; float fx, fy;
    clamp_cell(v.x, cx, fx);
    clamp_cell(v.y, cy, fy);
    const float q = v.z;
    const float gx = 1.0f - fx, gy = 1.0f - fy;
    float* p00 = &tile[(cx - x0) * TPAD + (cy - y0)];
    lds_fadd(p00,            gx * gy * q);  // (cx  , cy  )
    lds_fadd(p00 + 1,        gx * fy * q);  // (cx  , cy+1)
    lds_fadd(p00 + TPAD,     fx * gy * q);  // (cx+1, cy  )
    lds_fadd(p00 + TPAD + 1, fx * fy * q);  // (cx+1, cy+1)
  }

  // Inline-asm DS ops are invisible to the compiler's DScnt tracking: drain
  // them explicitly before the barrier that orders the LDS read-out.
  asm volatile("s_wait_dscnt 0x0" ::: "memory");
  __syncthreads();

  for (int idx = tid; idx < TPAD * TPAD; idx += K4_THREADS) {
    const int r = idx / TPAD, c = idx % TPAD;
    const int gx = x0 + r, gy = y0 + c;
    if (gx < CIC_GRID && gy < CIC_GRID) {
      const float val = tile[idx];
      const bool edge = (r == 0) | (c == 0) | (r == TILE) | (c == TILE);
      if (edge) {
        if (val != 0.0f)  // halo ring: shared with up to 3 neighbor tiles
          global_fadd_dev(out, ((unsigned)(gx * CIC_GRID + gy)) << 2, val);
      } else {
        out[gx * CIC_GRID + gy] = val;  // interior: exclusively ours
      }
    }
  }
}

// ---------------------------------------------------------------------------
// Fallback (ws too small): direct device-scope atomic deposition.
__global__ __launch_bounds__(256) void cic_direct_kernel(
    const f32x2* __restrict__ pos, const float* __restrict__ chg,
    float* __restrict__ out, int n) {
  const int stride = gridDim.x * blockDim.x;
  for (int i = blockIdx.x * blockDim.x + threadIdx.x; i < n; i += stride) {
    if (i + stride < n) {
      __builtin_prefetch((const void*)(pos + i + stride), 0, 0);
      __builtin_prefetch((const void*)(chg + i + stride), 0, 0);
    }
    f32x2 p = __builtin_nontemporal_load(pos + i);
    float q = __builtin_nontemporal_load(chg + i);
    int cx, cy; float fx, fy;
    clamp_cell(p.x, cx, fx);
    clamp_cell(p.y, cy, fy);
    unsigned base = ((unsigned)(cx * CIC_GRID + cy)) << 2;
    float gx = 1.0f - fx, gy = 1.0f - fy;
    global_fadd_dev(out, base,                      gx * gy * q);
    global_fadd_dev(out, base + 4u,                 gx * fy * q);
    global_fadd_dev(out, base + 4u * CIC_GRID,      fx * gy * q);
    global_fadd_dev(out, base + 4u * CIC_GRID + 4u, fx * fy * q);
  }
}

// ---------------------------------------------------------------------------
extern "C" void kernel_launch(void* const* d_in, const int* in_sizes, int n_in,
                              void* d_out, int out_size, void* d_ws, size_t ws_size,
                              hipStream_t stream) {
  const f32x2* pos = (const f32x2*)d_in[0];  // (N,2) f32
  const float* chg = (const float*)d_in[1];  // (N,)  f32
  float* out = (float*)d_out;                // 1024*1024 f32

  const int n  = in_sizes[1];
  const int n4 = out_size / 4;

  // Binned path needs: cursor header + 256 tile segments with headroom.
  // Uniform positions => per-tile count ~Poisson(n/256); slack is ~28 sigma.
  bool binned_path = false;
  size_t cap = 0;
  unsigned* cursor = nullptr;
  f32x4* binned = nullptr;
  if (d_ws && ws_size > WS_HDR) {
    cap = (ws_size - WS_HDR) / ((size_t)NBIN * sizeof(f32x4));
    size_t need = (size_t)n / NBIN + (size_t)n / 2048 + 1024;
    if (cap >= need) {
      binned_path = true;
      cursor = (unsigned*)d_ws;
      binned = (f32x4*)((char*)d_ws + WS_HDR);
    }
  }

  // 1) Re-zero output grid (+ cursors) every call: deterministic, poison-safe.
  zero_kernel<<<(n4 + 255) / 256, 256, 0, stream>>>((f32x4*)out, n4, cursor);

  if (binned_path) {
    const int per_block = K3_THREADS * K3_PPT;  // 4096 particles / block
    const int blocks3 = (n + per_block - 1) / per_block;
    scatter_kernel<<<blocks3, K3_THREADS, 0, stream>>>(pos, chg, binned, cursor,
                                                       n, (unsigned)cap);
    deposit_tile_kernel<<<NBIN, K4_THREADS, 0, stream>>>(binned, cursor, out,
                                                         (unsigned)cap);
  } else {
    cic_direct_kernel<<<4096, 256, 0, stream>>>(pos, chg, out, n);
  }
}